// ConvGen_40544491274893
// MI455X (gfx1250) — compile-verified
//
#include <hip/hip_runtime.h>

typedef __attribute__((ext_vector_type(2))) float v2f;
typedef __attribute__((ext_vector_type(8))) float v8f;

#define IC 256
#define OC 256
#define HH 56
#define WW 56
#define HWSZ (HH * WW)          // 3136
#define CHW (IC * HWSZ)         // stride between images
#define KTAPS 9
#define WROW (IC * KTAPS)       // 2304 = weight row length per oc

#define WAVE_OC 16
#define WAVE_POS 64             // 4 col-tiles of 16
#define BLOCK_OC 64             // 4 oc-subtiles
#define BLOCK_POS 128           // 2 pos-subtiles

__global__ __launch_bounds__(256) void conv3x3_wmma_f32(
    const float* __restrict__ x, const float* __restrict__ w,
    float* __restrict__ out)
{
  const int lane   = threadIdx.x & 31;
  const int wave   = threadIdx.x >> 5;
  const int l16    = lane & 15;
  const int hiHalf = lane >> 4;        // 0: lanes 0-15, 1: lanes 16-31
  const int kHi    = hiHalf * 2;       // K offset within 4-wide step (ISA 32-bit A/B layout)

  const int wave_oc  = wave & 3;
  const int wave_pos = wave >> 2;

  const int oc_base  = blockIdx.y * BLOCK_OC + wave_oc * WAVE_OC;
  const int pos_base = blockIdx.x * BLOCK_POS + wave_pos * WAVE_POS;

  const int m = oc_base + l16;         // A-matrix row (output channel) for this lane

  // Decode the 4 per-lane output positions (column tiles of 16)
  int n_[4], h_[4], w_[4], sp_[4];
#pragma unroll
  for (int j = 0; j < 4; ++j) {
    int p  = pos_base + j * 16 + l16;
    int n  = p / HWSZ;
    int sp = p - n * HWSZ;
    int hh = sp / WW;
    int ww = sp - hh * WW;
    n_[j] = n; h_[j] = hh; w_[j] = ww; sp_[j] = sp;
  }

  v8f c[4] = {};   // 4 accumulators: 16 oc x 64 positions

  const float* wRowBase = w + (long)m * WROW + kHi * KTAPS;

  for (int tap = 0; tap < KTAPS; ++tap) {
    const int dh = tap / 3 - 1;
    const int dw = tap % 3 - 1;

    // Per-tap shifted base pointers. Address is ALWAYS in-bounds (clamped to
    // plane start when the tap falls into padding); the loaded value is
    // zeroed with a data-select instead, so loads stay unconditional and
    // EXEC is never touched (WMMA requires EXEC all-1s anyway).
    const float* bptr[4];
    bool valid[4];
#pragma unroll
    for (int j = 0; j < 4; ++j) {
      int hh = h_[j] + dh, ww = w_[j] + dw;
      valid[j] = ((unsigned)hh < (unsigned)HH) && ((unsigned)ww < (unsigned)WW);
      long spOff = valid[j] ? (long)(hh * WW + ww) : 0L;   // clamp -> in-bounds
      bptr[j] = x + (long)n_[j] * CHW + (long)kHi * HWSZ + spOff;
    }

    const float* aptr = wRowBase + tap;

#pragma unroll 4
    for (int ic0 = 0; ic0 < IC; ic0 += 4) {
      // A frag: lane holds A[m][ic0+kHi], A[m][ic0+kHi+1]  (stride 9 floats in w)
      v2f a;
      a.x = aptr[0];
      a.y = aptr[KTAPS];
      aptr += 4 * KTAPS;

#pragma unroll
      for (int j = 0; j < 4; ++j) {
        // Unconditional loads from clamped address, then v_cndmask the data.
        float t0 = bptr[j][0];
        float t1 = bptr[j][HWSZ];
        bptr[j] += 4 * HWSZ;
        v2f b;
        b.x = valid[j] ? t0 : 0.0f;
        b.y = valid[j] ? t1 : 0.0f;

        c[j] = __builtin_amdgcn_wmma_f32_16x16x4_f32(
            /*neg_a=*/false, a, /*neg_b=*/false, b,
            /*c_mod=*/(short)0, c[j], /*reuse_a=*/false, /*reuse_b=*/false);
      }
    }
  }

  // Store: C layout -> VGPR r: M = r + 8*hiHalf, N = lane%16. Scale by 1/9.
  const float s = 1.0f / 9.0f;
#pragma unroll
  for (int j = 0; j < 4; ++j) {
    const int  ocr   = oc_base + hiHalf * 8;
    const long obase = (long)n_[j] * (long)OC * HWSZ + sp_[j];
#pragma unroll
    for (int r = 0; r < 8; ++r) {
      out[obase + (long)(ocr + r) * HWSZ] = c[j][r] * s;
    }
  }
}

extern "C" void kernel_launch(void* const* d_in, const int* in_sizes, int n_in,
                              void* d_out, int out_size, void* d_ws, size_t ws_size,
                              hipStream_t stream) {
  const float* x = (const float*)d_in[0];   // (32, 256, 56, 56) fp32
  const float* w = (const float*)d_in[1];   // (256, 256, 3, 3) fp32
  float* out = (float*)d_out;               // (32, 256, 56, 56) fp32

  // positions: 32*56*56 = 100352 = 784 * 128 ; channels: 256 = 4 * 64
  dim3 grid(100352 / BLOCK_POS, OC / BLOCK_OC, 1);
  dim3 block(256, 1, 1);
  conv3x3_wmma_f32<<<grid, block, 0, stream>>>(x, w, out);
}